// IF_19576460935541
// MI455X (gfx1250) — compile-verified
//
#include <hip/hip_runtime.h>
#include <hip/hip_bf16.h>
#include <math.h>

// Instantaneous-frequency from STFT phase.
//  - unwrap+diff collapses to a per-pair principal-value wrap: no cumsum needed.
//  - DFT done as fp32 GEMM on v_wmma_f32_16x16x4_f32 (full precision for atan2).
//  - One wave32 per block (EXEC all ones, as WMMA requires), 16 frames x 16 bins.
//  - Block-uniform interior/edge split keeps the hot K-loop branch-free.
//  - One __sincosf per iteration; second basis column via exact angle-addition.

#define N_FFTC   512
#define HOPC     128
#define PADSC    256
#define T_IN     160000
#define N_FRAMES 1251
#define N_BINS   257
#define BATCHC   32
#define OUT_ROWS_PER_TILE 15   // frame tiles overlap by 1 so diffs stay in-block

typedef __attribute__((ext_vector_type(2))) float v2f;
typedef __attribute__((ext_vector_type(8))) float v8f;

__device__ __forceinline__ int reflect_idx(int s) {
    s = (s < 0) ? -s : s;                               // left reflect (exclude edge)
    s = (s > T_IN - 1) ? (2 * (T_IN - 1) - s) : s;      // right reflect
    return s;
}

// K-loop: accumulate Sre = sum x*w*cos, Sim = sum x*w*sin on f32 WMMA.
// FAST=true -> all samples in-bounds, straight b64 loads (82/84 tiles).
template <bool FAST>
__device__ __forceinline__ void dft_accum(const float* __restrict__ wp, int base,
                                          int bin, int h,
                                          const float* __restrict__ win,
                                          v8f& cre, v8f& cim) {
    const float thstep = 3.14159265358979323846f / 256.0f;   // pi/256

    // Lane-constant rotation for the +1 K column: delta = bin*pi/256.
    float sb, cb;
    __sincosf(thstep * (float)bin, &sb, &cb);

    // ang = (bin * kk) mod 512, kk = k0 + 2h, k0 += 4 each iter (exact int recurrence).
    int ang  = (bin * 2 * h) & 511;
    const int step = (bin * 4) & 511;

    for (int k0 = 0; k0 < N_FFTC; k0 += 4) {
        const int kk = k0 + 2 * h;                      // this lane's K pair

        // ---- A operand: windowed frame samples (16x4 f32, 2 VGPRs) ----
        v2f a;
        if (FAST) {
            const float2 s = *(const float2*)(wp + base + kk);  // 8B aligned: kk even
            a.x = s.x; a.y = s.y;
        } else {
            a.x = wp[reflect_idx(base + kk)];
            a.y = wp[reflect_idx(base + kk + 1)];
        }
        const v2f w2 = *(const v2f*)&win[kk];            // ds_load_b64
        a.x *= w2.x; a.y *= w2.y;

        // ---- B operands: DFT basis (4x16 f32) ----
        const int t = ((ang + 256) & 511) - 256;         // centered exact reduction
        float s0, c0;
        __sincosf(thstep * (float)t, &s0, &c0);
        const float c1 = c0 * cb - s0 * sb;              // angle addition: theta + delta
        const float s1 = s0 * cb + c0 * sb;
        ang = (ang + step) & 511;

        v2f bc, bs;
        bc.x = c0; bc.y = c1;
        bs.x = s0; bs.y = s1;

        // D = A x B + C, fp32 WMMA
        cre = __builtin_amdgcn_wmma_f32_16x16x4_f32(false, a, false, bc, (short)0, cre, false, false);
        cim = __builtin_amdgcn_wmma_f32_16x16x4_f32(false, a, false, bs, (short)0, cim, false, false);
    }
}

__global__ __launch_bounds__(32) void if_phase_kernel(const float* __restrict__ wav,
                                                      float* __restrict__ out) {
    __shared__ float win[N_FFTC];
    __shared__ float ph[16 * 17];   // 16x16 phase tile, stride 17 vs bank conflicts

    const int lane = threadIdx.x;   // 0..31 (wave32)
    const int M    = lane & 15;     // A row / B col / C-D col within tile
    const int h    = lane >> 4;     // lane half: K pair select, C-D row half
    const int ft   = blockIdx.x;    // frame tile
    const int bt   = blockIdx.y;    // bin tile
    const int b    = blockIdx.z;    // batch

    // Periodic Hann window into LDS (512 values, 16 per lane).
    const float wstep = 6.283185307179586f / (float)N_FFTC;
    for (int j = lane; j < N_FFTC; j += 32)
        win[j] = 0.5f - 0.5f * __cosf(wstep * (float)j);
    __syncthreads();

    const int t0 = ft * OUT_ROWS_PER_TILE;
    int frame = t0 + M;
    if (frame > N_FRAMES - 1) frame = N_FRAMES - 1;     // clamp: dup rows never stored
    const int binBase = bt * 16;
    const int bin = binBase + M;                        // B-operand column for this lane

    const float* wp   = wav + (size_t)b * T_IN;
    const int    base = frame * HOPC - PADSC;           // sample index of n=0 (pre-reflect)

    v8f cre = {0.f,0.f,0.f,0.f,0.f,0.f,0.f,0.f};        // sum x*w*cos
    v8f cim = {0.f,0.f,0.f,0.f,0.f,0.f,0.f,0.f};        // sum x*w*sin  (im = -this)

    // Block-uniform interior test: every frame f in [t0, t0+15] has
    // f*HOP-PAD >= 0 and f*HOP-PAD+511 <= T_IN-1  <=>  t0 >= 2 && t0+15 <= 1248.
    if (t0 >= 2 && t0 + 15 <= 1248) {
        dft_accum<true >(wp, base, bin, h, win, cre, cim);   // scalar branch, no EXEC churn
    } else {
        dft_accum<false>(wp, base, bin, h, win, cre, cim);
    }

    // Phases -> LDS. C/D layout: VGPR j holds row (j + 8h), col = lane&15.
    #pragma unroll
    for (int j = 0; j < 8; ++j) {
        const int Mrow = j + 8 * h;
        ph[Mrow * 17 + M] = atan2f(-cim[j], cre[j]);
    }
    __syncthreads();

    // Wrapped frame-to-frame phase delta / pi  (== unwrap-then-diff of reference).
    const int colBin = binBase + M;
    if (colBin < N_BINS) {
        const float PIF     = 3.14159265358979323846f;
        const float TWO_PIF = 6.28318530717958647692f;
        const float INV_PI  = 0.31830988618379067154f;
        float* obase = out + (size_t)b * N_FRAMES * N_BINS + colBin;
        #pragma unroll
        for (int j = 0; j < 8; ++j) {
            const int Mrow = j + 8 * h;
            const int f = t0 + Mrow;
            if (Mrow == 0) {
                if (ft == 0) obase[0] = ph[M] * INV_PI;  // out[:,0,:] = phase0/pi
                continue;
            }
            if (f > N_FRAMES - 1) continue;
            const float p1v = ph[Mrow * 17 + M];
            const float p0v = ph[(Mrow - 1) * 17 + M];
            const float dd  = p1v - p0v;
            float m = fmodf(dd + PIF, TWO_PIF);
            if (m < 0.f) m += TWO_PIF;                   // jnp.mod semantics (sign of divisor)
            float ddmod = m - PIF;
            if (ddmod == -PIF && dd > 0.f) ddmod = PIF;  // edge fix from reference
            obase[(size_t)f * N_BINS] = ddmod * INV_PI;
        }
    }
}

extern "C" void kernel_launch(void* const* d_in, const int* in_sizes, int n_in,
                              void* d_out, int out_size, void* d_ws, size_t ws_size,
                              hipStream_t stream) {
    (void)in_sizes; (void)n_in; (void)out_size; (void)d_ws; (void)ws_size;
    const float* wav = (const float*)d_in[0];
    float* out = (float*)d_out;

    // frame tiles: outputs 1..1250 in chunks of 15 -> 84 tiles; bins: 17 tiles of 16.
    dim3 grid(84, 17, BATCHC);
    dim3 block(32);
    hipLaunchKernelGGL(if_phase_kernel, grid, block, 0, stream, wav, out);
}